// ConfidenceMarginLoss_64725157151246
// MI455X (gfx1250) — compile-verified
//
#include <hip/hip_runtime.h>
#include <hip/hip_bf16.h>
#include <stdint.h>

// ---------------------------------------------------------------------------
// ConfidenceMarginLoss, MI455X (gfx1250, wave32)
//   Kernel 1: one workgroup per row; single-pass online softmax producing
//             (max, sumexp, max-non-target) + target logit -> per-row loss.
//             Memory-bound: 824 MB / 23.3 TB/s ~= 35 us floor, single read.
//   Kernel 2: mean of 4096 row losses via V_WMMA_F32_16X16X4_F32 (exact f32),
//             ones(16x4) x data(4x16) accumulate, one wave.
// ---------------------------------------------------------------------------

typedef __attribute__((ext_vector_type(2))) float v2f;
typedef __attribute__((ext_vector_type(8))) float v8f;

#define BLOCK 512              // 16 waves of 32
#define NEG_BIG (-3.0e38f)     // finite "-inf" identity (avoids inf-inf NaNs)
#define EPS 1e-7f

// merge one scalar element into online (m, s) state and track non-target max
__device__ __forceinline__ void online_merge1(float x, int idx, int tgt,
                                              float& m, float& s, float& mnt) {
    float mn = fmaxf(m, x);
    s = s * __expf(m - mn) + __expf(x - mn);
    m = mn;
    if (idx != tgt) mnt = fmaxf(mnt, x);
}

__global__ __launch_bounds__(BLOCK) void cml_row_kernel(
        const float* __restrict__ inp, const int* __restrict__ tgt,
        float* __restrict__ row_loss, int C) {
    const int row = blockIdx.x;
    const int tid = threadIdx.x;
    const float* __restrict__ x = inp + (size_t)row * (size_t)C;
    const int t = tgt[row];

    float m   = NEG_BIG;   // running max (all classes)
    float s   = 0.0f;      // running sum of exp(x - m)
    float mnt = NEG_BIG;   // running max over non-target classes

    // --- alignment fixup: rows are only 4B-aligned (C = 50257 is odd) ---
    int head = (int)((16u - ((uint32_t)(uintptr_t)x & 15u)) & 15u) >> 2; // 0..3
    if (head > C) head = C;
    for (int i = tid; i < head; i += BLOCK)
        online_merge1(x[i], i, t, m, s, mnt);

    // --- main vectorized pass: global_load_b128 streams ---
    const int nvec = (C - head) >> 2;
    const float4* __restrict__ xv = (const float4*)(x + head);
    for (int i = tid; i < nvec; i += BLOCK) {
        float4 v = xv[i];
        const int base = head + (i << 2);
        // chunk max then 4 exps + 1 rescale exp (instead of 8)
        float cm = fmaxf(fmaxf(v.x, v.y), fmaxf(v.z, v.w));
        float mn = fmaxf(m, cm);
        float cs = __expf(v.x - mn) + __expf(v.y - mn) +
                   __expf(v.z - mn) + __expf(v.w - mn);
        s = s * __expf(m - mn) + cs;
        m = mn;
        // non-target max (mask the target column to -inf-equivalent)
        float a0 = (base + 0 == t) ? NEG_BIG : v.x;
        float a1 = (base + 1 == t) ? NEG_BIG : v.y;
        float a2 = (base + 2 == t) ? NEG_BIG : v.z;
        float a3 = (base + 3 == t) ? NEG_BIG : v.w;
        mnt = fmaxf(mnt, fmaxf(fmaxf(a0, a1), fmaxf(a2, a3)));
    }

    // --- scalar tail ---
    for (int i = head + (nvec << 2) + tid; i < C; i += BLOCK)
        online_merge1(x[i], i, t, m, s, mnt);

    // --- wave32 reduction (xor shuffle tree) ---
    #pragma unroll
    for (int off = 16; off >= 1; off >>= 1) {
        float m2   = __shfl_xor(m,   off, 32);
        float s2   = __shfl_xor(s,   off, 32);
        float mnt2 = __shfl_xor(mnt, off, 32);
        float mn = fmaxf(m, m2);
        s = s * __expf(m - mn) + s2 * __expf(m2 - mn);
        m = mn;
        mnt = fmaxf(mnt, mnt2);
    }

    // --- cross-wave combine via LDS (16 waves) ---
    __shared__ float sm[BLOCK / 32], ss[BLOCK / 32], st[BLOCK / 32];
    const int wave = tid >> 5, lane = tid & 31;
    if (lane == 0) { sm[wave] = m; ss[wave] = s; st[wave] = mnt; }
    __syncthreads();

    if (tid < 32) {
        const int nw = BLOCK / 32; // 16
        m   = (lane < nw) ? sm[lane] : NEG_BIG;
        s   = (lane < nw) ? ss[lane] : 0.0f;
        mnt = (lane < nw) ? st[lane] : NEG_BIG;
        #pragma unroll
        for (int off = 16; off >= 1; off >>= 1) {
            float m2   = __shfl_xor(m,   off, 32);
            float s2   = __shfl_xor(s,   off, 32);
            float mnt2 = __shfl_xor(mnt, off, 32);
            float mn = fmaxf(m, m2);
            s = s * __expf(m - mn) + s2 * __expf(m2 - mn);
            m = mn;
            mnt = fmaxf(mnt, mnt2);
        }
        if (lane == 0) {
            float xt    = x[t];
            float inv_s = 1.0f / s;
            float p_t   = __expf(xt  - m) * inv_s;     // target prob
            float p_mi  = __expf(mnt - m) * inv_s;     // max incorrect prob
            row_loss[row] = -__logf(p_t + EPS) - __logf(1.0f - p_mi + EPS);
        }
    }
}

// ---------------------------------------------------------------------------
// Mean of B row losses using V_WMMA_F32_16X16X4_F32 with an all-ones A.
// D[m][n] = sum_k B[k][n]; every row of D is identical, and C VGPR0 holds
// rows M=0 (lanes 0-15) and M=8 (lanes 16-31), so lane-sum(c[0]) = 2 * total.
// Exact f32 accumulation; fixed order -> deterministic.
// ---------------------------------------------------------------------------
__global__ __launch_bounds__(32) void cml_mean_kernel(
        const float* __restrict__ row_loss, float* __restrict__ out, int B) {
    const int lane = threadIdx.x;
    v2f a; a.x = 1.0f; a.y = 1.0f;     // A = ones(16x4), 2 VGPRs per lane
    v8f c = {};
    const int tiles = B >> 6;           // 64 f32 per WMMA tile (4x16 per call? -> 2 f32/lane)
    for (int tIdx = 0; tIdx < tiles; ++tIdx) {
        const float* p = row_loss + (tIdx << 6) + (lane << 1);
        v2f b = *(const v2f*)p;         // 8B-aligned global_load_b64
        c = __builtin_amdgcn_wmma_f32_16x16x4_f32(
                /*neg_a=*/false, a, /*neg_b=*/false, b,
                /*c_mod=*/(short)0, c, /*reuse_a=*/false, /*reuse_b=*/false);
    }
    float r = c[0] * 0.5f;              // lane-sum over c[0] double-counts total
    for (int i = (tiles << 6) + lane; i < B; i += 32)  // tail (B % 64)
        r += row_loss[i];
    #pragma unroll
    for (int off = 16; off >= 1; off >>= 1)
        r += __shfl_xor(r, off, 32);
    if (lane == 0) out[0] = r / (float)B;
}

extern "C" void kernel_launch(void* const* d_in, const int* in_sizes, int n_in,
                              void* d_out, int out_size, void* d_ws, size_t ws_size,
                              hipStream_t stream) {
    const float* inp = (const float*)d_in[0];
    const int*   tgt = (const int*)d_in[1];
    const int B = in_sizes[1];               // 4096
    const int C = in_sizes[0] / B;           // 50257
    float* row_loss = (float*)d_ws;          // B floats of scratch

    cml_row_kernel<<<B, BLOCK, 0, stream>>>(inp, tgt, row_loss, C);
    cml_mean_kernel<<<1, 32, 0, stream>>>(row_loss, (float*)d_out, B);
}